// HGNNEncoder_14087492731429
// MI455X (gfx1250) — compile-verified
//
#include <hip/hip_runtime.h>
#include <math.h>

// ---------------------------------------------------------------------------
// HGNN encoder pipeline for MI455X (gfx1250, wave32).
// All GEMMs run on V_WMMA_F32_16X16X4_F32 (full f32 precision, matrix cores),
// register-blocked 32x64 per wave (2x4 tiles): 8 WMMAs per K-step against
// 6 b64 fragment loads. GEMM memory traffic uses explicit address_space(1)
// pointers so loads lower to GLOBAL (LOADcnt only), not FLAT (LOADcnt+DScnt).
// ---------------------------------------------------------------------------

typedef float v2f __attribute__((ext_vector_type(2)));
typedef float v8f __attribute__((ext_vector_type(8)));

#if defined(__HIP_DEVICE_COMPILE__)
#define GLB __attribute__((address_space(1)))
#else
#define GLB
#endif

#define BB 128
#define NN 200
#define TT 512
#define HH 256
#define PHI_C   0.94247778f      // 3.1415926 * 0.3
#define MAXN_C  0.996f           // (1 - 4e-3)/sqrt(c), c = 1
#define ACT_A   0.5f             // all three activation couplings are 0.5

__device__ __forceinline__ v8f wmma_f32(v2f a, v2f b, v8f c)
{
    return __builtin_amdgcn_wmma_f32_16x16x4_f32(
               false, a, false, b, (short)0, c, false, false);
}

// ---------------------------------------------------------------------------
// Batched WMMA GEMM: C[b] = A[b] x B[b] (+ bias).
// One wave computes a 32x64 macro-tile (2x4 WMMA tiles, 8 accumulators).
//   A element [r,k] at  Ab + r*ar + k*ak     (ar/ak encode transposes)
//   B element [k,n] at  Bb + k*bk + n*bc
// K must be a multiple of 4 (true for 200/256/512/600/800).
// Out-of-range rows/cols are CLAMPED for loads (garbage only lands in C
// rows/cols the masked store never writes) -> straight-line inner loop.
// AK1/BK1: K-stride is compile-time 1 -> contiguous v2f (b64) fragment loads.
// ---------------------------------------------------------------------------
template<bool AK1, bool BK1>
__global__ __launch_bounds__(128)
void k_wmma_gemm(const float* __restrict__ A, size_t saB, int ar, int ak,
                 const float* __restrict__ Bm, size_t sbB, int bk, int bc,
                 const float* __restrict__ bias,
                 float* __restrict__ C, size_t scB, int ldc,
                 int M, int Nc, int K)
{
    const int lane = threadIdx.x & 31;
    const int wave = blockIdx.x * (blockDim.x >> 5) + (threadIdx.x >> 5);
    const int tN4 = (Nc + 63) >> 6;           // 64-wide column macro-tiles
    const int tM2 = (M + 31) >> 5;            // 32-tall row macro-tiles
    if (wave >= tM2 * tN4) return;            // wave-uniform: EXEC stays all-1s
    const int b  = blockIdx.y;
    const int m0 = (wave / tN4) << 5;
    const int n0 = (wave % tN4) << 6;

    const GLB float* Ab = (const GLB float*)(A  + (size_t)b * saB);
    const GLB float* Bb = (const GLB float*)(Bm + (size_t)b * sbB);

    // Fragment lane mapping (16x16x4 f32):
    //   A: lane holds row m = lane&15, K pair at k + 2*(lane>>4)
    //   B: lane holds col n = lane&15, same K pair
    const int l15 = lane & 15;
    const int kh  = (lane >> 4) << 1;

    const GLB float* ap[2];
    const GLB float* bp[4];
#pragma unroll
    for (int mi = 0; mi < 2; ++mi) {
        const int am = min(m0 + 16 * mi + l15, M - 1);
        ap[mi] = Ab + (size_t)am * ar + (size_t)kh * (AK1 ? 1 : ak);
    }
#pragma unroll
    for (int ni = 0; ni < 4; ++ni) {
        const int bn = min(n0 + 16 * ni + l15, Nc - 1);
        bp[ni] = Bb + (size_t)kh * (BK1 ? 1 : bk) + (size_t)bn * bc;
    }
    const size_t stepA = 4u * (size_t)(AK1 ? 1 : ak);
    const size_t stepB = 4u * (size_t)(BK1 ? 1 : bk);

    v8f acc[2][4];
#pragma unroll
    for (int mi = 0; mi < 2; ++mi)
#pragma unroll
        for (int ni = 0; ni < 4; ++ni)
            acc[mi][ni] = (v8f){0.f,0.f,0.f,0.f,0.f,0.f,0.f,0.f};

#pragma unroll 2
    for (int k = 0; k < K; k += 4) {
        v2f af[2], bf[4];
#pragma unroll
        for (int mi = 0; mi < 2; ++mi) {
            if (AK1) {
                af[mi] = *(const GLB v2f*)ap[mi];         // global_load_b64
            } else {
                af[mi].x = ap[mi][0];
                af[mi].y = ap[mi][ak];
            }
            ap[mi] += stepA;
        }
#pragma unroll
        for (int ni = 0; ni < 4; ++ni) {
            if (BK1) {
                bf[ni] = *(const GLB v2f*)bp[ni];         // global_load_b64
            } else {
                bf[ni].x = bp[ni][0];
                bf[ni].y = bp[ni][bk];
            }
            bp[ni] += stepB;
        }
#pragma unroll
        for (int mi = 0; mi < 2; ++mi)
#pragma unroll
            for (int ni = 0; ni < 4; ++ni)
                acc[mi][ni] = wmma_f32(af[mi], bf[ni], acc[mi][ni]);
    }

    // C/D layout: lane -> col (lane&15), rows i + 8*(lane>>4), i = 0..7
    GLB float* Cb = (GLB float*)(C + (size_t)b * scB);
    const GLB float* biasg = (const GLB float*)bias;
    const int rb = m0 + ((lane >> 4) << 3);
#pragma unroll
    for (int ni = 0; ni < 4; ++ni) {
        const int cn = n0 + 16 * ni + l15;
        if (cn >= Nc) continue;
        const float bd = bias ? biasg[cn] : 0.f;
#pragma unroll
        for (int mi = 0; mi < 2; ++mi) {
#pragma unroll
            for (int i = 0; i < 8; ++i) {
                const int r = rb + 16 * mi + i;
                if (r < M) Cb[(size_t)r * ldc + cn] = acc[mi][ni][i] + bd;
            }
        }
    }
}

// ---------------------------------------------------------------------------
// norm_adj(|A|): D^-1/2 (|A| + I) D^-1/2, one block per batch.
// ---------------------------------------------------------------------------
__global__ __launch_bounds__(256)
void k_norm_adj(const float* __restrict__ A, float* __restrict__ O)
{
    const int b = blockIdx.x;
    const float* a = A + (size_t)b * NN * NN;
    float*       o = O + (size_t)b * NN * NN;
    __shared__ float dsh[NN];
    for (int i = threadIdx.x; i < NN; i += blockDim.x) {
        float s = 1.0f;                         // + identity on the diagonal
        for (int j = 0; j < NN; ++j) s += fabsf(a[(size_t)i * NN + j]);
        dsh[i] = rsqrtf(s);
    }
    __syncthreads();
    for (int idx = threadIdx.x; idx < NN * NN; idx += blockDim.x) {
        const int i = idx / NN, j = idx % NN;
        const float v = fabsf(a[idx]) + ((i == j) ? 1.0f : 0.f);
        o[idx] = v * dsh[i] * dsh[j];
    }
}

// ---------------------------------------------------------------------------
// fkernel (project + logmap0, c=1) fused with act = relu(x) + a*cos(x+phi).
// One block per row, blockDim == H == 256.
// ---------------------------------------------------------------------------
__global__ __launch_bounds__(256)
void k_fk_act(const float* __restrict__ X, float* __restrict__ Y)
{
    const size_t row = (size_t)blockIdx.y * NN + blockIdx.x;
    const int tid = threadIdx.x;
    const float v = X[row * HH + tid];
    __shared__ float sh[256];
    sh[tid] = v * v;
    __syncthreads();
    for (int s = 128; s > 0; s >>= 1) {
        if (tid < s) sh[tid] += sh[tid + s];
        __syncthreads();
    }
    const float nr  = sqrtf(sh[0]);
    const float n1  = fmaxf(nr, 1e-15f);
    const float spr = (n1 > MAXN_C) ? (MAXN_C / n1) : 1.0f;   // project scale
    const float n2  = fmaxf(nr * spr, 1e-15f);                // projected norm
    const float t   = fminf(n2, 1.0f - 1e-7f);
    const float fac = atanhf(t) / n2 * spr;                   // logmap0∘project
    const float x1  = fac * v;
    Y[row * HH + tid] = fmaxf(x1, 0.f) + ACT_A * cosf(x1 + PHI_C);
}

// Row normalize: Y = X / (||X|| + eps), blockDim == H == 256.
__global__ __launch_bounds__(256)
void k_rownorm(const float* __restrict__ X, float* __restrict__ Y, float eps)
{
    const size_t row = (size_t)blockIdx.y * NN + blockIdx.x;
    const int tid = threadIdx.x;
    const float v = X[row * HH + tid];
    __shared__ float sh[256];
    sh[tid] = v * v;
    __syncthreads();
    for (int s = 128; s > 0; s >>= 1) {
        if (tid < s) sh[tid] += sh[tid + s];
        __syncthreads();
    }
    Y[row * HH + tid] = v / (sqrtf(sh[0]) + eps);
}

// Column means of fMRI [B,T,N] over T.
__global__ void k_colmean(const float* __restrict__ X, float* __restrict__ mu)
{
    const int i = blockIdx.x * blockDim.x + threadIdx.x;
    const int b = blockIdx.y;
    if (i >= NN) return;
    const float* x = X + (size_t)b * TT * NN + i;
    float s = 0.f;
    for (int t = 0; t < TT; ++t) s += x[(size_t)t * NN];
    mu[b * NN + i] = s * (1.0f / TT);
}

// s[b,i] = sqrt(max(G_ii - T*mu_i^2, 0))
__global__ void k_corr_diag(const float* __restrict__ G,
                            const float* __restrict__ mu,
                            float* __restrict__ sd)
{
    const int b = blockIdx.x;
    for (int i = threadIdx.x; i < NN; i += blockDim.x) {
        const float m = mu[b * NN + i];
        const float c = G[((size_t)b * NN + i) * NN + i] - (float)TT * m * m;
        sd[b * NN + i] = sqrtf(fmaxf(c, 0.f));
    }
}

// In place: G <- |nan_to_num((G - T mu_i mu_j)/(s_i s_j))|
__global__ void k_corr_fin(float* __restrict__ G,
                           const float* __restrict__ mu,
                           const float* __restrict__ sd)
{
    const size_t idx = (size_t)blockIdx.x * blockDim.x + threadIdx.x;
    const size_t total = (size_t)BB * NN * NN;
    if (idx >= total) return;
    const int    j = (int)(idx % NN);
    const size_t r = idx / NN;
    const int    i = (int)(r % NN);
    const int    b = (int)(r / NN);
    const float cov = G[idx] - (float)TT * mu[b * NN + i] * mu[b * NN + j];
    const float den = sd[b * NN + i] * sd[b * NN + j];
    float v = (den > 0.f) ? fabsf(cov / den) : 0.f;       // 0/0 -> nan -> 0
    G[idx] = fminf(v, 3.4028235e38f);                     // inf -> max float
}

// f = concat(0.5*DTI/(||.||+1e-8), 0.5*adj_fMRI/(||.||+1e-8)) -> [B,N,800]
__global__ __launch_bounds__(256)
void k_build_f(const float* __restrict__ DTI, const float* __restrict__ aF,
               float* __restrict__ f)
{
    const int b = blockIdx.y, n = blockIdx.x, tid = threadIdx.x;
    const float* drow = DTI + ((size_t)b * NN + n) * (3 * NN);
    const float* frow = aF  + ((size_t)b * NN + n) * NN;
    float*       orow = f   + ((size_t)b * NN + n) * (4 * NN);
    __shared__ float sh[256];

    float p = 0.f;
    for (int i = tid; i < 3 * NN; i += 256) { const float x = drow[i]; p += x * x; }
    sh[tid] = p; __syncthreads();
    for (int s = 128; s > 0; s >>= 1) { if (tid < s) sh[tid] += sh[tid + s]; __syncthreads(); }
    const float invd = 0.5f / (sqrtf(sh[0]) + 1e-8f);
    __syncthreads();

    p = 0.f;
    for (int i = tid; i < NN; i += 256) { const float x = frow[i]; p += x * x; }
    sh[tid] = p; __syncthreads();
    for (int s = 128; s > 0; s >>= 1) { if (tid < s) sh[tid] += sh[tid + s]; __syncthreads(); }
    const float invf = 0.5f / (sqrtf(sh[0]) + 1e-8f);

    for (int i = tid; i < 3 * NN; i += 256) orow[i]          = drow[i] * invd;
    for (int i = tid; i < NN;     i += 256) orow[3 * NN + i] = frow[i] * invf;
}

// ---------------------------------------------------------------------------
static inline void gemm(hipStream_t stream,
                        const float* A, size_t saB, int ar, int ak,
                        const float* Bm, size_t sbB, int bk, int bc,
                        const float* bias, float* C, size_t scB, int ldc,
                        int M, int Nc, int K)
{
    const int tiles = ((M + 31) / 32) * ((Nc + 63) / 64);
    dim3 grid((tiles + 3) / 4, BB);
    if (ak == 1 && bk == 1)
        k_wmma_gemm<true, true><<<grid, 128, 0, stream>>>(
            A, saB, ar, ak, Bm, sbB, bk, bc, bias, C, scB, ldc, M, Nc, K);
    else if (ak == 1)
        k_wmma_gemm<true, false><<<grid, 128, 0, stream>>>(
            A, saB, ar, ak, Bm, sbB, bk, bc, bias, C, scB, ldc, M, Nc, K);
    else if (bk == 1)
        k_wmma_gemm<false, true><<<grid, 128, 0, stream>>>(
            A, saB, ar, ak, Bm, sbB, bk, bc, bias, C, scB, ldc, M, Nc, K);
    else
        k_wmma_gemm<false, false><<<grid, 128, 0, stream>>>(
            A, saB, ar, ak, Bm, sbB, bk, bc, bias, C, scB, ldc, M, Nc, K);
}

extern "C" void kernel_launch(void* const* d_in, const int* in_sizes, int n_in,
                              void* d_out, int out_size, void* d_ws, size_t ws_size,
                              hipStream_t stream)
{
    (void)in_sizes; (void)n_in; (void)out_size; (void)ws_size;

    const float* DTI     = (const float*)d_in[0];
    const float* adjDTIi = (const float*)d_in[1];
    const float* fMRI    = (const float*)d_in[2];
    const float* Wd1 = (const float*)d_in[3];  const float* bd1 = (const float*)d_in[4];
    const float* Wd2 = (const float*)d_in[5];  const float* bd2 = (const float*)d_in[6];
    const float* Wf1 = (const float*)d_in[7];  const float* bf1 = (const float*)d_in[8];
    const float* Wf2 = (const float*)d_in[9];  const float* bf2 = (const float*)d_in[10];
    const float* Wh1 = (const float*)d_in[11]; const float* bh1 = (const float*)d_in[12];
    const float* Wh2 = (const float*)d_in[13]; const float* bh2 = (const float*)d_in[14];

    float* out    = (float*)d_out;
    float* o_fea  = out;                 // [B,N,H]
    float* o_adjc = out + 6553600;       // [B,N,N]
    float* o_dD   = out + 11673600;      // [B,N,H]
    float* o_dF   = out + 18227200;      // [B,N,H]
    float* o_aD   = out + 24780800;      // [B,N,N] passthrough adj_DTI
    float* o_aF   = out + 29900800;      // [B,N,N] adj_fMRI

    float* ws    = (float*)d_ws;
    float* w_adj = ws;                   // 5,120,000  (adjD -> adjF -> adjC)
    float* w_T   = ws + 5120000;         // 6,553,600
    float* w_U   = w_T + 6553600;        // 6,553,600
    float* w_f   = w_U + 6553600;        // 20,480,000 [B,N,800]
    float* w_mu  = w_f + 20480000;       // 25,600
    float* w_s   = w_mu + 25600;         // 25,600

    const size_t sNN = (size_t)NN * NN;   // 40,000
    const size_t sNH = (size_t)NN * HH;   // 51,200

    // adj_DTI passthrough
    hipMemcpyAsync(o_aD, adjDTIi, (size_t)BB * sNN * sizeof(float),
                   hipMemcpyDeviceToDevice, stream);

    // ---------------- DTI branch ----------------
    k_norm_adj<<<BB, 256, 0, stream>>>(adjDTIi, w_adj);                       // adjD
    gemm(stream, DTI,  (size_t)NN*3*NN, 3*NN, 1, Wd1, 0, HH, 1, nullptr,
         w_T, sNH, HH, NN, HH, 3*NN);                                         // T = DTI@Wd1
    gemm(stream, w_adj, sNN, NN, 1, w_T, sNH, HH, 1, bd1,
         w_U, sNH, HH, NN, HH, NN);                                           // U = adjD@T + bd1
    k_fk_act<<<dim3(NN, BB), 256, 0, stream>>>(w_U, w_T);                     // g1
    gemm(stream, w_T, sNH, HH, 1, Wd2, 0, HH, 1, nullptr,
         w_U, sNH, HH, NN, HH, HH);                                           // T2 = g1@Wd2
    gemm(stream, w_adj, sNN, NN, 1, w_U, sNH, HH, 1, bd2,
         w_T, sNH, HH, NN, HH, NN);                                           // U2
    k_fk_act<<<dim3(NN, BB), 256, 0, stream>>>(w_T, o_dD);                    // data_DTI

    // ---------------- fMRI correlation ----------------
    k_colmean<<<dim3(1, BB), 256, 0, stream>>>(fMRI, w_mu);
    gemm(stream, fMRI, (size_t)TT*NN, 1, NN,      // A = X^T  (A[i,k] = X[k,i])
                 fMRI, (size_t)TT*NN, NN, 1,      // B = X
         nullptr, o_aF, sNN, NN, NN, NN, TT);                                 // G = X^T X
    k_corr_diag<<<BB, 256, 0, stream>>>(o_aF, w_mu, w_s);
    k_corr_fin<<<(int)((BB*sNN + 255) / 256), 256, 0, stream>>>(o_aF, w_mu, w_s);
    k_norm_adj<<<BB, 256, 0, stream>>>(o_aF, w_adj);                          // adjF

    // ---------------- fMRI branch (features = adj_fMRI) ----------------
    gemm(stream, o_aF, sNN, NN, 1, Wf1, 0, HH, 1, nullptr,
         w_T, sNH, HH, NN, HH, NN);
    gemm(stream, w_adj, sNN, NN, 1, w_T, sNH, HH, 1, bf1,
         w_U, sNH, HH, NN, HH, NN);
    k_fk_act<<<dim3(NN, BB), 256, 0, stream>>>(w_U, w_T);
    gemm(stream, w_T, sNH, HH, 1, Wf2, 0, HH, 1, nullptr,
         w_U, sNH, HH, NN, HH, HH);
    gemm(stream, w_adj, sNN, NN, 1, w_U, sNH, HH, 1, bf2,
         w_T, sNH, HH, NN, HH, NN);
    k_fk_act<<<dim3(NN, BB), 256, 0, stream>>>(w_T, o_dF);                    // data_fMRI

    // ---------------- coupling ----------------
    k_rownorm<<<dim3(NN, BB), 256, 0, stream>>>(o_dF, w_T, 0.f);              // Fn
    k_rownorm<<<dim3(NN, BB), 256, 0, stream>>>(o_dD, w_U, 1e-7f);            // Dn
    gemm(stream, w_T, sNH, HH, 1,                 // A = Fn [N,H]
         w_U, sNH, 1, HH,                         // B = Dn^T (B[k,n] = Dn[n,k])
         nullptr, o_adjc, sNN, NN, NN, NN, HH);                               // adj_coupled
    k_norm_adj<<<BB, 256, 0, stream>>>(o_adjc, w_adj);                        // adjC

    // ---------------- HGNN branch ----------------
    k_build_f<<<dim3(NN, BB), 256, 0, stream>>>(DTI, o_aF, w_f);              // f [B,N,800]
    gemm(stream, w_f, (size_t)NN*4*NN, 4*NN, 1, Wh1, 0, HH, 1, nullptr,
         w_T, sNH, HH, NN, HH, 4*NN);
    gemm(stream, w_adj, sNN, NN, 1, w_T, sNH, HH, 1, bh1,
         w_U, sNH, HH, NN, HH, NN);
    k_fk_act<<<dim3(NN, BB), 256, 0, stream>>>(w_U, w_T);
    gemm(stream, w_T, sNH, HH, 1, Wh2, 0, HH, 1, nullptr,
         w_U, sNH, HH, NN, HH, HH);
    gemm(stream, w_adj, sNN, NN, 1, w_U, sNH, HH, 1, bh2,
         w_T, sNH, HH, NN, HH, NN);
    k_fk_act<<<dim3(NN, BB), 256, 0, stream>>>(w_T, o_fea);                   // fea_coupled
}